// BatchedDynamicEmbeddingTables_31825707663710
// MI455X (gfx1250) — compile-verified
//
#include <hip/hip_runtime.h>

// BatchedDynamicEmbeddingTables: gather + SUM pool, F=2 tables, B=4096, L=20, D=128.
// HBM-bound (~88 MB moved / call -> ~3.8 us floor at 23.3 TB/s). One wave32 per
// (feature, batch) output row. Rows staged global->LDS with CDNA5 async loads
// (ASYNCcnt-tracked), double buffered, reduced from LDS, stored as 512B/wave.

#define V_CAP   1000000
#define D_DIM   128
#define B_BATCH 4096
#define L_BAG   20
#define F_NUM   2

#define CHUNK   5              // rows per async chunk (L_BAG = 4 chunks of 5)
#define WAVES_PER_BLOCK 4

// per-wave LDS: 2 buffers x CHUNK rows x 512B; float4 granularity (32 lanes x 16B)
#define WAVE_F4   (2 * CHUNK * 32)        // float4 elements per wave
#define BUF_F4    (CHUNK * 32)            // float4 elements per buffer

__global__ __launch_bounds__(WAVES_PER_BLOCK * 32)
void emb_pool_async_kernel(const int* __restrict__ idx,
                           const float* __restrict__ t0,
                           const float* __restrict__ t1,
                           float* __restrict__ out) {
    __shared__ float4 lbuf[WAVES_PER_BLOCK * WAVE_F4];   // 20 KB / block

    const int lane = threadIdx.x & 31;
    // make the wave id an SGPR so the bag-index loads become scalar loads
    const int wave = __builtin_amdgcn_readfirstlane((int)(threadIdx.x >> 5));
    const int gw   = blockIdx.x * WAVES_PER_BLOCK + wave;   // 0 .. F*B-1
    const int f    = gw & 1;
    const int b    = gw >> 1;

    const float* tab = f ? t1 : t0;
    const int*   ip  = idx + ((size_t)(f * B_BATCH + b) * L_BAG);

    // wave-uniform bag indices -> scalar registers
    int rows[L_BAG];
#pragma unroll
    for (int l = 0; l < L_BAG; ++l) rows[l] = ip[l];

    // per-wave LDS region; addresses for the async engine are derived from the
    // real pointer so lbuf's address escapes into the asm (keeps the ds_load
    // side alive) and the low 32 bits of the flat address are the LDS offset.
    float4* const wbase = &lbuf[wave * WAVE_F4];

    // issue one chunk: CHUNK async row copies, each = wave32 x 16B = 512B row
    auto issue_chunk = [&](int c, int bufsel) {
#pragma unroll
        for (int r = 0; r < CHUNK; ++r) {
            const float* g = tab + (size_t)rows[c * CHUNK + r] * D_DIM + (lane << 2);
            float4* dst = wbase + bufsel * BUF_F4 + r * 32 + lane;
            const unsigned laddr = (unsigned)(uintptr_t)dst;   // LDS byte offset
            asm volatile("global_load_async_to_lds_b128 %0, %1, off"
                         :: "v"(laddr), "v"(g)
                         : "memory");
        }
    };

    float4 acc;
    acc.x = 0.f; acc.y = 0.f; acc.z = 0.f; acc.w = 0.f;

    auto accum = [&](int bufsel) {
        const float4* p = wbase + bufsel * BUF_F4 + lane;
#pragma unroll
        for (int r = 0; r < CHUNK; ++r) {
            float4 v = p[r * 32];
            acc.x += v.x; acc.y += v.y; acc.z += v.z; acc.w += v.w;
        }
        // ensure our LDS reads retired before the async engine may overwrite
        asm volatile("s_wait_dscnt 0x0" ::: "memory");
    };

    // 4-chunk double-buffered pipeline (async completions are in-order)
    issue_chunk(0, 0);                                   // ASYNCcnt = 5
    issue_chunk(1, 1);                                   // ASYNCcnt = 10
    asm volatile("s_wait_asynccnt 0x5" ::: "memory");    // chunk0 landed
    accum(0);
    issue_chunk(2, 0);
    asm volatile("s_wait_asynccnt 0x5" ::: "memory");    // chunk1 landed
    accum(1);
    issue_chunk(3, 1);
    asm volatile("s_wait_asynccnt 0x5" ::: "memory");    // chunk2 landed
    accum(0);
    asm volatile("s_wait_asynccnt 0x0" ::: "memory");    // chunk3 landed
    accum(1);

    // out[b, f*128 + lane*4 .. +3]  (512B coalesced store per wave)
    float4* orow = (float4*)(out + (size_t)b * (F_NUM * D_DIM) + f * D_DIM);
    orow[lane] = acc;
}

extern "C" void kernel_launch(void* const* d_in, const int* in_sizes, int n_in,
                              void* d_out, int out_size, void* d_ws, size_t ws_size,
                              hipStream_t stream) {
    (void)in_sizes; (void)n_in; (void)out_size; (void)d_ws; (void)ws_size;
    const int*   idx = (const int*)d_in[0];    // [F, B, L] int32
    const float* t0  = (const float*)d_in[1];  // [V, D] fp32
    const float* t1  = (const float*)d_in[2];  // [V, D] fp32
    float*       out = (float*)d_out;          // [B, F*D] fp32

    const int total_waves = F_NUM * B_BATCH;                    // 8192
    const int blocks = total_waves / WAVES_PER_BLOCK;           // 2048
    emb_pool_async_kernel<<<blocks, WAVES_PER_BLOCK * 32, 0, stream>>>(idx, t0, t1, out);
}